// SubTokenEmbedding_6734508720780
// MI455X (gfx1250) — compile-verified
//
#include <hip/hip_runtime.h>
#include <hip/hip_bf16.h>
#include <stdint.h>

// ---------------------------------------------------------------------------
// SubTokenEmbedding: fused embedding gather + sorted-segment sum pooling.
// Bandwidth-bound (~230 MB total -> ~10us @ 23.3 TB/s). No matrix structure,
// so no WMMA; CDNA5 paths: per-wave TDM tensor_load_to_lds + s_wait_tensorcnt,
// LDS broadcast reads, global_prefetch_b8, NT stores for interior rows,
// global_atomic_add_f32 only at chunk boundaries.
// ---------------------------------------------------------------------------

#define H_EMB 128
#define H4 (H_EMB / 4)            // 32 float4 per embedding row
#define WAVE_SIZE 32
#define WAVES_PER_BLOCK 8
#define BLOCK_THREADS (WAVE_SIZE * WAVES_PER_BLOCK)          // 256
#define CHUNK_PER_WAVE 64
#define CHUNK_PER_BLOCK (CHUNK_PER_WAVE * WAVES_PER_BLOCK)   // 512

#if __has_builtin(__builtin_amdgcn_tensor_load_to_lds) && __has_builtin(__builtin_amdgcn_s_wait_tensorcnt)
#define HAVE_TDM 1
#endif

typedef unsigned int u32;
typedef u32 u32x4 __attribute__((ext_vector_type(4)));
typedef int i32x4 __attribute__((ext_vector_type(4)));
typedef int i32x8 __attribute__((ext_vector_type(8)));
typedef float f32x4 __attribute__((ext_vector_type(4)));   // native vector: OK for
                                                           // __builtin_nontemporal_store

#ifdef HAVE_TDM
// Build a D# for a 1-D tile of 32-bit elements and issue TENSOR_LOAD_TO_LDS.
// Descriptor layout per CDNA5 ISA §8: group0 {count, lds_addr, global_addr,
// type=2}; group1 {data_size=4B, tensor_dim0=avail (OOB tile elements read as
// zero and are never dereferenced), tile_dim0}.
__device__ __forceinline__ void tdm_load_1d_to_lds(u32 lds_addr, const void* gsrc,
                                                   u32 avail_elems, u32 tile_elems) {
  uint64_t ga = (uint64_t)(uintptr_t)gsrc;
  u32x4 g0;
  g0.x = 1u;                                            // count=1, user mode
  g0.y = lds_addr;                                      // lds_addr (bytes)
  g0.z = (u32)ga;                                       // global_addr[31:0]
  g0.w = (u32)((ga >> 32) & 0x01FFFFFFu) | (2u << 30);  // global_addr[56:32], type=2
  i32x8 g1;
  g1[0] = (int)(2u << 16);                              // data_size=4B; wg mask=0
  g1[1] = (int)((avail_elems & 0xFFFFu) << 16);         // tensor_dim0[15:0] @ bits 63:48
  g1[2] = (int)((avail_elems >> 16) | (1u << 16));      // tensor_dim0[31:16]; tensor_dim1=1
  g1[3] = (int)(tile_elems << 16);                      // tile_dim0 @ bits 127:112
  g1[4] = 0;                                            // tile_dim1/2 unused
  g1[5] = (int)tile_elems;                              // tensor_dim0_stride
  g1[6] = 0;
  g1[7] = 0;
  i32x4 gz = {0, 0, 0, 0};
#if __clang_major__ >= 23
  i32x8 gz8 = {0, 0, 0, 0, 0, 0, 0, 0};
  __builtin_amdgcn_tensor_load_to_lds(g0, g1, gz, gz, gz8, 0);
#else
  __builtin_amdgcn_tensor_load_to_lds(g0, g1, gz, gz, 0);
#endif
}
#endif  // HAVE_TDM

__global__ __launch_bounds__(BLOCK_THREADS) void zero_out_kernel(f32x4* __restrict__ out,
                                                                 int n4) {
  int i = blockIdx.x * blockDim.x + threadIdx.x;
  int stride = gridDim.x * blockDim.x;
  f32x4 z = {0.0f, 0.0f, 0.0f, 0.0f};
  for (; i < n4; i += stride) out[i] = z;
}

__device__ __forceinline__ void flush_run(float* __restrict__ out, int seg, int lane,
                                          f32x4 v, bool use_atomic) {
  float* p = out + (size_t)seg * H_EMB + lane * 4;
  if (use_atomic) {
    // Boundary run: row may be shared with neighbouring chunks -> L2 atomics.
    atomicAdd(p + 0, v.x);
    atomicAdd(p + 1, v.y);
    atomicAdd(p + 2, v.z);
    atomicAdd(p + 3, v.w);
  } else {
    // Interior run: this wave is the exclusive owner; written once, never
    // re-read -> non-temporal store keeps L2 for table/ids/boundary rows.
    __builtin_nontemporal_store(v, (f32x4*)p);
  }
}

__global__ __launch_bounds__(BLOCK_THREADS) void seg_embed_sum_kernel(
    const float* __restrict__ table,  // [VOCAB, 128]
    const int* __restrict__ ids,      // [n_sub]
    const int* __restrict__ segs,     // [n_sub], sorted
    float* __restrict__ out,          // [n_nodes, 128], pre-zeroed
    int n_sub) {
  __shared__ int sIds[CHUNK_PER_BLOCK];
  __shared__ int sSegs[CHUNK_PER_BLOCK];

  const int base = blockIdx.x * CHUNK_PER_BLOCK;
  const int wave = threadIdx.x / WAVE_SIZE;  // 0..7, each wave owns a 64-chunk
  const int lane = threadIdx.x % WAVE_SIZE;  // lane l -> columns 4l..4l+3
  const int off = wave * CHUNK_PER_WAVE;
  const int wbase = base + off;

  int cnt = n_sub - wbase;
  if (cnt > CHUNK_PER_WAVE) cnt = CHUNK_PER_WAVE;

#ifdef HAVE_TDM
  // Per-wave DMA of this wave's own id/seg chunk into its own LDS region.
  // TDM ignores EXEC and issues once per wave reaching the instruction, so
  // there is no cross-wave dependency and no barrier needed. avail is clamped
  // to [0, CHUNK_PER_WAVE]; OOB tile elements are descriptor-bounded (zero).
  u32 avail = (u32)(cnt > 0 ? cnt : 0);
  tdm_load_1d_to_lds((u32)(uintptr_t)(&sIds[off]), ids + wbase, avail, CHUNK_PER_WAVE);
  tdm_load_1d_to_lds((u32)(uintptr_t)(&sSegs[off]), segs + wbase, avail, CHUNK_PER_WAVE);
  __builtin_amdgcn_s_wait_tensorcnt(0);
  asm volatile("" ::: "memory");  // keep LDS reads below the tensor wait
  if (cnt <= 0) return;
#else
  for (int i = threadIdx.x; i < CHUNK_PER_BLOCK; i += BLOCK_THREADS) {
    int g = base + i;
    sIds[i] = (g < n_sub) ? ids[g] : 0;
    sSegs[i] = (g < n_sub) ? segs[g] : 0;
  }
  __syncthreads();
  if (cnt <= 0) return;
#endif

  const f32x4* __restrict__ t4 = (const f32x4*)table;

  f32x4 acc = {0.0f, 0.0f, 0.0f, 0.0f};
  int run_seg = sSegs[off];      // LDS broadcast read (all lanes same addr)
  bool touches_start = true;     // first run may continue from previous chunk

  for (int i = 0; i < cnt; ++i) {
    int seg = sSegs[off + i];
    if (seg != run_seg) {
      flush_run(out, run_seg, lane, acc, touches_start);
      acc = (f32x4){0.0f, 0.0f, 0.0f, 0.0f};
      run_seg = seg;
      touches_start = false;
    }
    int id = sIds[off + i];
    if (i + 4 < cnt) {
      int id_next = sIds[off + i + 4];
      __builtin_prefetch(&t4[(size_t)id_next * H4 + lane], 0, 1);  // global_prefetch_b8
    }
    f32x4 r = t4[(size_t)id * H4 + lane];  // global_load_b128, wave covers full row
    acc += r;
  }
  // Last run may continue into the next chunk -> atomic.
  flush_run(out, run_seg, lane, acc, true);
}

extern "C" void kernel_launch(void* const* d_in, const int* in_sizes, int n_in,
                              void* d_out, int out_size, void* d_ws, size_t ws_size,
                              hipStream_t stream) {
  const float* table = (const float*)d_in[0];
  const int* ids = (const int*)d_in[1];
  const int* segs = (const int*)d_in[2];
  float* out = (float*)d_out;
  const int n_sub = in_sizes[1];

  // 1) zero the output (empty segments must be 0; d_out is poisoned)
  int n4 = out_size / 4;
  int zgrid = (n4 + BLOCK_THREADS - 1) / BLOCK_THREADS;
  if (zgrid > 2048) zgrid = 2048;
  zero_out_kernel<<<zgrid, BLOCK_THREADS, 0, stream>>>((f32x4*)d_out, n4);

  // 2) gather + sorted-segment sum
  int blocks = (n_sub + CHUNK_PER_BLOCK - 1) / CHUNK_PER_BLOCK;
  seg_embed_sum_kernel<<<blocks, BLOCK_THREADS, 0, stream>>>(table, ids, segs, out, n_sub);
}